// ModifiedBarlowTwinsLoss_36120674959540
// MI455X (gfx1250) — compile-verified
//
#include <hip/hip_runtime.h>
#include <hip/hip_bf16.h>
#include <math.h>

typedef __attribute__((ext_vector_type(2))) float v2f;
typedef __attribute__((ext_vector_type(8))) float v8f;

#define NROWS 8192
#define DDIM  512

// ---------------------------------------------------------------------------
// Kernel 1: per-column stats for both matrices, folded into affine coefs:
//   norm(x) = (x - mean) / std * (1/sqrt(D))  ==  x*mul[k] + add[k]
// coef layout (floats): [mulA:512][addA:512][mulB:512][addB:512]
// ---------------------------------------------------------------------------
__global__ __launch_bounds__(256) void colstats_kernel(
    const float* __restrict__ zA, const float* __restrict__ zB,
    float* __restrict__ coef)
{
    const int mat = blockIdx.y;                 // 0 = A, 1 = B
    const float* __restrict__ src = mat ? zB : zA;
    float* __restrict__ mul = coef + mat * 1024;
    float* __restrict__ add = coef + mat * 1024 + 512;

    const int t   = threadIdx.x;
    const int col = blockIdx.x * 32 + (t & 31); // 32 cols per block, coalesced
    const int rl  = t >> 5;                     // 8 row-groups

    double s = 0.0, s2 = 0.0;
    for (int r = rl; r < NROWS; r += 8) {
        float x = src[(size_t)r * DDIM + col];
        s  += (double)x;
        s2 += (double)x * (double)x;
    }

    __shared__ double rs[256];
    __shared__ double rq[256];
    rs[t] = s; rq[t] = s2;
    __syncthreads();

    if (rl == 0) {
        #pragma unroll
        for (int j = 1; j < 8; ++j) { s += rs[t + j * 32]; s2 += rq[t + j * 32]; }
        double mean = s / (double)NROWS;
        double var  = (s2 - s * s / (double)NROWS) / (double)(NROWS - 1); // ddof=1
        float istd  = (float)(1.0 / sqrt(var));
        const float sc = 0.044194173824159216f;  // 1/sqrt(512)  (folds the /d)
        float m = istd * sc;
        mul[col] = m;
        add[col] = (float)(-mean) * m;
    }
}

// ---------------------------------------------------------------------------
// Kernel 2: fused GEMM (c = Anorm * Bnorm^T, scaling pre-folded) + loss tile.
// Block tile 128x128, BK=32, 8 waves (wave32) each owning 64x32 via 4x2 WMMA
// f32 16x16x4 accumulators. Normalization applied as one FMA while staging
// global -> LDS. Epilogue computes sum((c - m)^2) per block.
// ---------------------------------------------------------------------------
#define LDSS 36   // LDS row stride in floats: 144B -> float4-aligned, no bank conflicts

__global__ __launch_bounds__(256) void gemm_loss_kernel(
    const float* __restrict__ zA, const float* __restrict__ zB,
    const int*  __restrict__ labels,
    const float* __restrict__ coef,
    float* __restrict__ partials)
{
    const float* __restrict__ mulA = coef;
    const float* __restrict__ addA = coef + 512;
    const float* __restrict__ mulB = coef + 1024;
    const float* __restrict__ addB = coef + 1536;

    __shared__ float As[128 * LDSS];
    __shared__ float Bs[128 * LDSS];
    __shared__ int   labR[128];
    __shared__ int   labC[128];
    __shared__ float red[256];

    const int t    = threadIdx.x;
    const int lane = t & 31;
    const int w    = t >> 5;        // wave 0..7
    const int wm   = w >> 2;        // 0..1 : 64-row stripe
    const int wn   = w & 3;         // 0..3 : 32-col stripe
    const int bM   = blockIdx.y * 128;
    const int bN   = blockIdx.x * 128;

    if (t < 128) {
        labR[t] = labels[bM + t];
        labC[t] = labels[bN + t];
    }

    const v8f z8 = {0.f, 0.f, 0.f, 0.f, 0.f, 0.f, 0.f, 0.f};
    v8f acc[4][2];
    #pragma unroll
    for (int mi = 0; mi < 4; ++mi)
        #pragma unroll
        for (int ni = 0; ni < 2; ++ni) acc[mi][ni] = z8;

    for (int s = 0; s < DDIM / 32; ++s) {
        const int kk = s * 32;
        __syncthreads();   // previous-stage readers done
        #pragma unroll
        for (int i = 0; i < 4; ++i) {
            int slot = i * 256 + t;           // 1024 float4 slots per tile
            int row  = slot >> 3;             // 0..127
            int c4   = (slot & 7) << 2;       // k offset within stage (0,4,..28)
            int k    = kk + c4;

            float4 ca = *(const float4*)(mulA + k);
            float4 aa = *(const float4*)(addA + k);
            float4 xa = *(const float4*)(zA + (size_t)(bM + row) * DDIM + k);
            float4 ya;
            ya.x = fmaf(xa.x, ca.x, aa.x);
            ya.y = fmaf(xa.y, ca.y, aa.y);
            ya.z = fmaf(xa.z, ca.z, aa.z);
            ya.w = fmaf(xa.w, ca.w, aa.w);
            *(float4*)&As[row * LDSS + c4] = ya;

            float4 cb = *(const float4*)(mulB + k);
            float4 ab = *(const float4*)(addB + k);
            float4 xb = *(const float4*)(zB + (size_t)(bN + row) * DDIM + k);
            float4 yb;
            yb.x = fmaf(xb.x, cb.x, ab.x);
            yb.y = fmaf(xb.y, cb.y, ab.y);
            yb.z = fmaf(xb.z, cb.z, ab.z);
            yb.w = fmaf(xb.w, cb.w, ab.w);
            *(float4*)&Bs[row * LDSS + c4] = yb;
        }
        __syncthreads();   // stage visible

        #pragma unroll
        for (int ks = 0; ks < 8; ++ks) {
            // ISA f32 16x16x4 A layout: lane holds M=(lane&15), K-pair=(lane>>4)*2+{0,1}
            const int koff = ks * 4 + ((lane >> 4) << 1);
            v2f a[4], b[2];
            #pragma unroll
            for (int mi = 0; mi < 4; ++mi) {
                int r = wm * 64 + mi * 16 + (lane & 15);
                a[mi] = *(const v2f*)&As[r * LDSS + koff];
            }
            #pragma unroll
            for (int ni = 0; ni < 2; ++ni) {
                int r = wn * 32 + ni * 16 + (lane & 15);
                b[ni] = *(const v2f*)&Bs[r * LDSS + koff];
            }
            #pragma unroll
            for (int mi = 0; mi < 4; ++mi)
                #pragma unroll
                for (int ni = 0; ni < 2; ++ni)
                    acc[mi][ni] = __builtin_amdgcn_wmma_f32_16x16x4_f32(
                        false, a[mi], false, b[ni],
                        (short)0, acc[mi][ni], false, false);
        }
    }

    // Epilogue: c already == Anorm.Bnorm^T/d due to pre-scaling; fuse (c-m)^2.
    float local = 0.f;
    #pragma unroll
    for (int mi = 0; mi < 4; ++mi)
        #pragma unroll
        for (int ni = 0; ni < 2; ++ni)
            #pragma unroll
            for (int v = 0; v < 8; ++v) {
                int r = wm * 64 + mi * 16 + v + ((lane >> 4) << 3); // M = v + 8*(lane/16)
                int c = wn * 32 + ni * 16 + (lane & 15);            // N = lane%16
                float m = (labR[r] == labC[c]) ? 1.f : 0.f;
                float d = acc[mi][ni][v] - m;
                local = fmaf(d, d, local);
            }

    red[t] = local;
    __syncthreads();
    #pragma unroll
    for (int off = 128; off > 0; off >>= 1) {
        if (t < off) red[t] += red[t + off];
        __syncthreads();
    }
    if (t == 0) partials[blockIdx.y * gridDim.x + blockIdx.x] = red[0];
}

// ---------------------------------------------------------------------------
// Kernel 3: deterministic final reduction (fixed order, no float atomics).
// ---------------------------------------------------------------------------
__global__ __launch_bounds__(256) void final_reduce_kernel(
    const float* __restrict__ partials, int n, float* __restrict__ out)
{
    __shared__ double red[256];
    const int t = threadIdx.x;
    double s = 0.0;
    for (int i = t; i < n; i += 256) s += (double)partials[i];
    red[t] = s;
    __syncthreads();
    #pragma unroll
    for (int off = 128; off > 0; off >>= 1) {
        if (t < off) red[t] += red[t + off];
        __syncthreads();
    }
    if (t == 0) out[0] = (float)red[0];
}

// ---------------------------------------------------------------------------
extern "C" void kernel_launch(void* const* d_in, const int* in_sizes, int n_in,
                              void* d_out, int out_size, void* d_ws, size_t ws_size,
                              hipStream_t stream)
{
    const float* zA     = (const float*)d_in[0];
    const float* zB     = (const float*)d_in[1];
    const int*   labels = (const int*)d_in[2];
    float*       out    = (float*)d_out;

    float* coef     = (float*)d_ws;   // 2048 floats of coefficients
    float* partials = coef + 2048;    // 64*64 = 4096 block partials

    colstats_kernel<<<dim3(16, 2), 256, 0, stream>>>(zA, zB, coef);
    gemm_loss_kernel<<<dim3(64, 64), 256, 0, stream>>>(zA, zB, labels, coef, partials);
    final_reduce_kernel<<<1, 256, 0, stream>>>(partials, 64 * 64, out);
}